// CondConv2d_32083405701663
// MI455X (gfx1250) — compile-verified
//
#include <hip/hip_runtime.h>
#include <hip/hip_bf16.h>

typedef __attribute__((ext_vector_type(2))) float v2f;
typedef __attribute__((ext_vector_type(8))) float v8f;

#define CIN   256
#define COUT  256
#define NEXP  8
#define HH    56
#define WW    56
#define NPIX  (HH * WW)        // 3136 = 49 * 64
#define KTOT  (CIN * 9)        // 2304
#define NSAMP 32
#define KCHUNKS 576            // KTOT / 4
#define HP    58               // padded H
#define WP    58               // padded W
#define PPIX  (HP * WP)        // 3364
#define NCP   (CIN / 2)        // 128 channel pairs

// ---------------------------------------------------------------------------
// Kernel 1: gate[n][e] = sigmoid( mean_hw(x[n]) . fc_w[e] + fc_b[e] )
// ---------------------------------------------------------------------------
__global__ void gate_kernel(const float* __restrict__ x,
                            const float* __restrict__ fc_w,
                            const float* __restrict__ fc_b,
                            float* __restrict__ gate) {
  __shared__ float m[CIN];
  const int n    = blockIdx.x;
  const int tid  = threadIdx.x;
  const int wave = tid >> 5;
  const int lane = tid & 31;
  const float* xb = x + (size_t)n * CIN * NPIX;

  for (int i = 0; i < 32; ++i) {
    const int c = wave * 32 + i;
    const float* xc = xb + (size_t)c * NPIX;
    float s = 0.f;
    for (int j = lane; j < NPIX; j += 32) s += xc[j];
    for (int off = 16; off > 0; off >>= 1) s += __shfl_down(s, off, 32);
    if (lane == 0) m[c] = s * (1.0f / NPIX);
  }
  __syncthreads();

  if (tid < NEXP) {
    float l = fc_b[tid];
    const float* w = fc_w + (size_t)tid * CIN;
    for (int c = 0; c < CIN; ++c) l += m[c] * w[c];
    gate[n * NEXP + tid] = 1.f / (1.f + __expf(-l));
  }
}

// ---------------------------------------------------------------------------
// Kernel 2: mixed weights written directly in WMMA A-fragment order.
// K-order tap-major: kk = tap*256 + ci.
//   wmix[ ((n*16 + cb)*576 + ks)*64 + lane*2 + j ]
//     = Wmix[n][cout = cb*16 + (lane&15)][kk = ks*4 + (lane>>4)*2 + j]
// ---------------------------------------------------------------------------
__global__ void mix_kernel(const float* __restrict__ wbank,   // (E*COUT, CIN, 3, 3)
                           const float* __restrict__ gate,    // (N, E)
                           float* __restrict__ wmix) {
  long t = (long)blockIdx.x * blockDim.x + threadIdx.x;  // 32*16*576*32 threads
  const int lane = (int)(t & 31);
  long r = t >> 5;
  const int ks = (int)(r % KCHUNKS); r /= KCHUNKS;
  const int cb = (int)(r % 16);
  const int n  = (int)(r / 16);

  const int co  = cb * 16 + (lane & 15);
  const int kk  = ks * 4 + ((lane >> 4) << 1);
  const int tap = kk >> 8;
  const int ci  = kk & 255;

  float g[NEXP];
#pragma unroll
  for (int e = 0; e < NEXP; ++e) g[e] = gate[n * NEXP + e];

  float s0 = 0.f, s1 = 0.f;
#pragma unroll
  for (int e = 0; e < NEXP; ++e) {
    const float* wp = wbank + ((((size_t)e * COUT + co) * CIN + ci) * 9 + tap);
    s0 += g[e] * wp[0];
    s1 += g[e] * wp[9];          // ci+1 -> +9 floats
  }
  ((float2*)wmix)[(((size_t)n * 16 + cb) * KCHUNKS + ks) * 32 + lane] =
      make_float2(s0, s1);
}

// ---------------------------------------------------------------------------
// Kernel 3: zero-pad + channel-pair interleave:
//   xt[n][cpair][ihp][iwp][2]  (HP=WP=58, pad=1)
// ---------------------------------------------------------------------------
__global__ void pad_kernel(const float* __restrict__ x,
                           float* __restrict__ xt) {
  const long idx = (long)blockIdx.x * blockDim.x + threadIdx.x; // 32*128*3364
  const int  iw  = (int)(idx % WP);
  long r = idx / WP;
  const int  ih  = (int)(r % HP); r /= HP;
  const int  cp  = (int)(r % NCP);
  const int  n   = (int)(r / NCP);

  const bool inb = (ih >= 1) && (ih <= HH) && (iw >= 1) && (iw <= WW);
  float v0 = 0.f, v1 = 0.f;
  if (inb) {
    const float* p = x + (((size_t)n * CIN + cp * 2) * HH + (ih - 1)) * WW + (iw - 1);
    v0 = p[0];
    v1 = p[NPIX];
  }
  ((float2*)xt)[idx] = make_float2(v0, v1);
}

// ---------------------------------------------------------------------------
// Kernel 4: implicit-GEMM conv.  Block = 128 cout x 64 pixel tile, 8 waves;
// wave = 32 cout x 32 pixel (4 f32 accumulators, 4 WMMAs per K=4 chunk).
// B slabs (32 cpairs x 64 pixels = 16 KB) are staged with the CDNA5 async
// global->LDS path (no VGPR round-trip, tracked by ASYNCcnt); A streams as
// coalesced b64 from the pre-swizzled wmix.
// ---------------------------------------------------------------------------
__global__ void __launch_bounds__(256)
conv_kernel(const float* __restrict__ xt,
            const float* __restrict__ wmix,
            float* __restrict__ out) {
  __shared__ float2 Bs[32 * 64];               // [cpairLocal][pix] : 16 KB

  const int ptile = blockIdx.x;  // 0..48
  const int ctile = blockIdx.y;  // 0..1   (128-cout tiles)
  const int n     = blockIdx.z;  // 0..31
  const int tid   = threadIdx.x;
  const int lane  = tid & 31;
  const int wave  = tid >> 5;
  const int cb2   = wave & 3;    // 32-cout sub-block
  const int ph    = wave >> 2;   // 32-pixel half

  const int cbG0 = ctile * 8 + cb2 * 2;        // first 16-cout block (0..15)
  // A-fragment stream: this lane's 2 floats; 2nd cout block at +36864 floats
  const float* ap = wmix + (((size_t)n * 16 + cbG0) * KCHUNKS) * 64 + lane * 2;

  const float* xtn = xt + (size_t)n * NCP * PPIX * 2;
  const int pixL   = ph * 32 + (lane & 15);    // pixel lane column (sub 0)
  const int laneh  = lane >> 4;                // 0: K+0/1, 1: K+2/3

  v8f acc00 = {}, acc01 = {}, acc10 = {}, acc11 = {};

  for (int kb = 0; kb < 36; ++kb) {            // K in slabs of 64
    const int tap = kb >> 2;                   // 4 slabs per tap
    const int cpb = (kb & 3) * 32;             // cpair base within tap
    const int dhp = tap / 3;                   // padded row offset (oh + dhp)
    const int dwp = tap % 3;

    __syncthreads();
    // async stage: 2048 float2, 8 per thread, direct global->LDS (ASYNCcnt)
#pragma unroll
    for (int i = 0; i < 8; ++i) {
      const int idx = tid + i * 256;
      const int cpl = idx >> 6;
      const int pix = idx & 63;
      const int p   = ptile * 64 + pix;
      const int ihp = p / WW + dhp;
      const int iwp = p % WW + dwp;
      const float2* gptr =
          (const float2*)(xtn + ((size_t)(cpb + cpl) * PPIX + ihp * WP + iwp) * 2);
      const unsigned ldsAddr = (unsigned)(uintptr_t)(&Bs[idx]);  // addr[31:0]
      asm volatile("global_load_async_to_lds_b64 %0, %1, off"
                   :
                   : "v"(ldsAddr), "v"(gptr)
                   : "memory");
    }
    asm volatile("s_wait_asynccnt 0x0" ::: "memory");
    __syncthreads();

#pragma unroll
    for (int cc = 0; cc < 16; ++cc) {          // 16 K=4 chunks per slab
      v2f a0 = *(const v2f*)ap;
      v2f a1 = *(const v2f*)(ap + 36864);      // +576*64 floats (147456 B imm)
      ap += 64;

      const float2 t0 = Bs[(cc * 2 + laneh) * 64 + pixL];
      const float2 t1 = Bs[(cc * 2 + laneh) * 64 + pixL + 16];
      v2f b0; b0.x = t0.x; b0.y = t0.y;
      v2f b1; b1.x = t1.x; b1.y = t1.y;

      acc00 = __builtin_amdgcn_wmma_f32_16x16x4_f32(false, a0, false, b0, (short)0, acc00, false, false);
      acc01 = __builtin_amdgcn_wmma_f32_16x16x4_f32(false, a0, false, b1, (short)0, acc01, false, false);
      acc10 = __builtin_amdgcn_wmma_f32_16x16x4_f32(false, a1, false, b0, (short)0, acc10, false, false);
      acc11 = __builtin_amdgcn_wmma_f32_16x16x4_f32(false, a1, false, b1, (short)0, acc11, false, false);
    }
  }

  // D layout: VGPR r -> M = r + 8*(lane>=16), N = lane&15
  float* ob = out + (size_t)n * COUT * NPIX;
  const int cout0 = cbG0 * 16 + (laneh << 3);
  const int px0   = ptile * 64 + pixL;
#pragma unroll
  for (int r = 0; r < 8; ++r) {
    ob[(size_t)(cout0 + r) * NPIX + px0]           = acc00[r];
    ob[(size_t)(cout0 + r) * NPIX + px0 + 16]      = acc01[r];
    ob[(size_t)(cout0 + 16 + r) * NPIX + px0]      = acc10[r];
    ob[(size_t)(cout0 + 16 + r) * NPIX + px0 + 16] = acc11[r];
  }
}

// ---------------------------------------------------------------------------
extern "C" void kernel_launch(void* const* d_in, const int* in_sizes, int n_in,
                              void* d_out, int out_size, void* d_ws, size_t ws_size,
                              hipStream_t stream) {
  const float* x     = (const float*)d_in[0];  // (32, 256, 56, 56)
  const float* wbank = (const float*)d_in[1];  // (2048, 256, 3, 3)
  const float* fc_w  = (const float*)d_in[2];  // (8, 256)
  const float* fc_b  = (const float*)d_in[3];  // (8,)
  float* out  = (float*)d_out;                 // (32, 256, 56, 56)

  float* gate = (float*)d_ws;                                   // 1 KB
  float* wmix = (float*)((char*)d_ws + 1024);                   // 75.5 MB
  float* xt   = (float*)((char*)d_ws + 1024 + (size_t)NSAMP * 16 * KCHUNKS * 64 * 4);

  gate_kernel<<<NSAMP, 256, 0, stream>>>(x, fc_w, fc_b, gate);

  const long mixThreads = (long)NSAMP * 16 * KCHUNKS * 32;      // 9,437,184
  mix_kernel<<<(int)(mixThreads / 256), 256, 0, stream>>>(wbank, gate, wmix);

  const long padThreads = (long)NSAMP * NCP * PPIX;             // 13,778,944
  pad_kernel<<<(int)(padThreads / 256), 256, 0, stream>>>(x, xt);

  dim3 grid(NPIX / 64, COUT / 128, NSAMP);  // (49, 2, 32)
  conv_kernel<<<grid, 256, 0, stream>>>(xt, wmix, out);
}